// TaskAlignedAssigner_53575422050327
// MI455X (gfx1250) — compile-verified
//
#include <hip/hip_runtime.h>
#include <cstdint>
#include <cstddef>

#define BB   8
#define NN   16800
#define MM   96
#define CC   80
#define KK   10
#define TPB  256
#define EPSF    1e-9f
#define IOUEPSF 1e-7f

// ---------------- CDNA5 async global->LDS helpers (ASYNCcnt path) ----------------
__device__ __forceinline__ void async_g2l_b32(const void* lds_p, const void* gp) {
  unsigned dst = (unsigned)(unsigned long long)lds_p;   // low 32 bits = LDS offset
  unsigned long long ga = (unsigned long long)gp;
  asm volatile("global_load_async_to_lds_b32 %0, %1, off"
               :: "v"(dst), "v"(ga) : "memory");
}
__device__ __forceinline__ void wait_async0() {
  asm volatile("s_wait_asynccnt 0" ::: "memory");
}

// ---------------- CDNA5 Tensor Data Mover (TENSORcnt path) ----------------
#if __has_builtin(__builtin_amdgcn_tensor_load_to_lds)
#define HAVE_TDM 1
typedef unsigned int u32x4 __attribute__((ext_vector_type(4)));
typedef int          i32x8 __attribute__((ext_vector_type(8)));
typedef int          i32x4 __attribute__((ext_vector_type(4)));

// 1-D tile copy: n_dwords DWORDs from global -> LDS, per ISA D# group0/1 layout.
// 6-arg builtin form (clang-23 / therock-10.0 lane).
__device__ __forceinline__ void tdm_copy_1d(const void* lds_p, const void* gp,
                                            unsigned n_dwords) {
  unsigned lds_addr = (unsigned)(unsigned long long)lds_p;
  unsigned long long ga = (unsigned long long)gp;
  u32x4 g0;
  g0[0] = 1u;                                           // count=1, is_restore=0, gather off
  g0[1] = lds_addr;                                     // lds_addr
  g0[2] = (unsigned)(ga & 0xFFFFFFFFull);               // global_addr[31:0]
  g0[3] = (unsigned)((ga >> 32) & 0x01FFFFFFull) | (2u << 30);  // addr[56:32] | type=2
  i32x8 g1;
  g1[0] = (int)(2u << 16);                              // wg_mask=0, data_size=2 (4B)
  g1[1] = (int)((n_dwords & 0xFFFFu) << 16);            // tensor_dim0[15:0] @bits63:48
  g1[2] = (int)((n_dwords >> 16) | (1u << 16));         // tensor_dim0 hi | tensor_dim1=1
  g1[3] = (int)((n_dwords & 0xFFFFu) << 16);            // tile_dim0 @bits127:112
  g1[4] = 0;                                            // tile_dim1=0 (1-D), tile_dim2=0
  g1[5] = (int)n_dwords;                                // tensor_dim0_stride[31:0]
  g1[6] = 0;
  g1[7] = 0;
  i32x4 gz4 = {0, 0, 0, 0};
  i32x8 gz8 = {0, 0, 0, 0, 0, 0, 0, 0};
  __builtin_amdgcn_tensor_load_to_lds(g0, g1, gz4, gz4, gz8, 0);
}
#else
#define HAVE_TDM 0
#endif

// ---------------- small helpers ----------------
__device__ __forceinline__ float iou_clip(float px1, float py1, float px2, float py2,
                                          float gx1, float gy1, float gx2, float gy2,
                                          float garea) {
  float ix1 = fmaxf(px1, gx1), iy1 = fmaxf(py1, gy1);
  float ix2 = fminf(px2, gx2), iy2 = fminf(py2, gy2);
  float iw = fmaxf(ix2 - ix1, 0.f), ih = fmaxf(iy2 - iy1, 0.f);
  float inter = iw * ih;
  float pa = (px2 - px1) * (py2 - py1);
  float iou = inter / (pa + garea - inter + IOUEPSF);
  return fmaxf(iou, 0.f);
}

// ---------------- kernel 0: zero the topk membership bitmask ----------------
__global__ void taa_zero_bits(unsigned* __restrict__ p, int count) {
  int i = blockIdx.x * blockDim.x + threadIdx.x;
  if (i < count) p[i] = 0u;
}

// ---------------- kernel 1: per (b,m) top-10 over N, scatter bits ----------------
__global__ void taa_topk_kernel(const float* __restrict__ pd_scores,
                                const float* __restrict__ pd_bboxes,
                                const int*   __restrict__ gt_labels,
                                const float* __restrict__ gt_bboxes,
                                const float* __restrict__ mask_gt,
                                float*       __restrict__ maxpg,      // [B*M]
                                unsigned*    __restrict__ maskbits)   // [B*N*3]
{
  const int bm = blockIdx.x;
  const int b  = bm / MM;
  const int m  = bm % MM;
  const int t  = threadIdx.x;

  // uniform (scalar) GT data
  const float gx1 = gt_bboxes[bm * 4 + 0];
  const float gy1 = gt_bboxes[bm * 4 + 1];
  const float gx2 = gt_bboxes[bm * 4 + 2];
  const float gy2 = gt_bboxes[bm * 4 + 3];
  const float ga  = (gx2 - gx1) * (gy2 - gy1);
  const float mg  = mask_gt[bm];
  const int   gl  = gt_labels[bm];

  // per-lane register-resident sorted top-10: (value desc, index asc)
  float tv[KK]; int ti[KK];
#pragma unroll
  for (int k = 0; k < KK; ++k) { tv[k] = -1e38f; ti[k] = 0x7FFFFFFF; }

  const float4* pb4 = reinterpret_cast<const float4*>(pd_bboxes) + (size_t)b * NN;
  const float*  srow = pd_scores + (size_t)b * NN * CC;

  for (int n = t; n < NN; n += TPB) {
    float4 pb = pb4[n];
    float iou = iou_clip(pb.x, pb.y, pb.z, pb.w, gx1, gy1, gx2, gy2, ga);
    float i2 = iou * iou;
    float s = srow[(size_t)n * CC + gl];
    float sig = 1.f / (1.f + expf(-s));
    float cv = sig * (i2 * i2 * i2) * mg;
    int ci = n;
#pragma unroll
    for (int k = 0; k < KK; ++k) {
      bool better = (cv > tv[k]) || (cv == tv[k] && ci < ti[k]);
      if (better) { float ov = tv[k]; int oi = ti[k]; tv[k] = cv; ti[k] = ci; cv = ov; ci = oi; }
    }
  }

  __shared__ float sv[TPB * KK];
  __shared__ int   si[TPB * KK];
#pragma unroll
  for (int k = 0; k < KK; ++k) { sv[t * KK + k] = tv[k]; si[t * KK + k] = ti[k]; }

  // log2(256)=8 step LDS tree merge of sorted-10 lists, (value desc, index asc)
  for (int stride = TPB / 2; stride >= 1; stride >>= 1) {
    __syncthreads();
    if (t < stride) {
      float av[KK]; int ai[KK];
#pragma unroll
      for (int k = 0; k < KK; ++k) { av[k] = sv[t * KK + k]; ai[k] = si[t * KK + k]; }
#pragma unroll
      for (int k = 0; k < KK; ++k) {
        float cv = sv[(t + stride) * KK + k];
        int   ci = si[(t + stride) * KK + k];
#pragma unroll
        for (int j = 0; j < KK; ++j) {
          bool better = (cv > av[j]) || (cv == av[j] && ci < ai[j]);
          if (better) { float ov = av[j]; int oi = ai[j]; av[j] = cv; ai[j] = ci; cv = ov; ci = oi; }
        }
      }
#pragma unroll
      for (int k = 0; k < KK; ++k) { sv[t * KK + k] = av[k]; si[t * KK + k] = ai[k]; }
    }
  }
  __syncthreads();

  float maxv = sv[0];
  if (t == 0) maxpg[bm] = fmaxf(maxv, EPSF);
  if (maxv > EPSF && t < KK) {  // keep flag -> scatter 10 bits in parallel
    atomicOr(&maskbits[((size_t)b * NN + si[t]) * 3 + (m >> 5)], 1u << (m & 31));
  }
}

// ---------------- kernel 2: per-anchor assignment ----------------
__global__ void taa_assign_kernel(const float* __restrict__ pd_scores,
                                  const float* __restrict__ pd_bboxes,
                                  const float* __restrict__ anchor_points,
                                  const int*   __restrict__ gt_labels,
                                  const float* __restrict__ gt_bboxes,
                                  const float* __restrict__ mask_gt,
                                  const float* __restrict__ maxpg,
                                  const unsigned* __restrict__ maskbits,
                                  float* __restrict__ out_bb,
                                  float* __restrict__ out_sc,
                                  float* __restrict__ out_fg,
                                  float* __restrict__ out_ti)
{
  const int b  = blockIdx.y;
  const int n0 = blockIdx.x * TPB;
  const int t  = threadIdx.x;
  const int n  = n0 + t;
  const bool valid = (n < NN);
  const int nc = valid ? n : (NN - 1);

  __shared__ float4 sgt4[MM];      // gt boxes, b128-readable
  __shared__ int    sgl[MM];
  __shared__ float  smg[MM];
  __shared__ float  smx[MM];
  float* sgt = (float*)sgt4;

  // ---- stage gt boxes via Tensor Data Mover (one wave issues the DMA) ----
#if HAVE_TDM
  if (t < 32) {
    tdm_copy_1d(sgt, gt_bboxes + (size_t)b * MM * 4, MM * 4);
  }
#else
  for (int j = t; j < MM * 4; j += TPB)
    async_g2l_b32(&sgt[j], gt_bboxes + (size_t)b * MM * 4 + j);
#endif

  // ---- stage labels / mask / per-gt max via async global->LDS (288 dwords) ----
  for (int j = t; j < 3 * MM; j += TPB) {
    const void* src; const void* dst;
    if (j < MM)          { src = gt_labels + (size_t)b * MM + j;            dst = &sgl[j]; }
    else if (j < 2 * MM) { src = mask_gt   + (size_t)b * MM + (j - MM);     dst = &smg[j - MM]; }
    else                 { src = maxpg     + (size_t)b * MM + (j - 2 * MM); dst = &smx[j - 2 * MM]; }
    async_g2l_b32(dst, src);
  }
  wait_async0();
#if HAVE_TDM
  if (t < 32) __builtin_amdgcn_s_wait_tensorcnt(0);
#endif
  __syncthreads();

  // ---- per-anchor data ----
  const float4 pb = reinterpret_cast<const float4*>(pd_bboxes)[(size_t)b * NN + nc];
  const float ax = anchor_points[(size_t)nc * 2 + 0];
  const float ay = anchor_points[(size_t)nc * 2 + 1];
  const unsigned* mbp = maskbits + ((size_t)b * NN + nc) * 3;
  const unsigned mb0 = mbp[0], mb1 = mbp[1], mb2 = mbp[2];
  const float* srow = pd_scores + ((size_t)b * NN + nc) * CC;

  float ovbest = -1e30f; int ovarg = 0;
  int fgc = 0, firstm = -1;
  float na = 0.f;

  for (int m = 0; m < MM; ++m) {
    float4 g = sgt4[m];                                // ds_load_b128
    float garea = (g.z - g.x) * (g.w - g.y);

    bool in_gts = (ax - g.x > EPSF) && (ay - g.y > EPSF) &&
                  (g.z - ax > EPSF) && (g.w - ay > EPSF);

    float iou = iou_clip(pb.x, pb.y, pb.z, pb.w, g.x, g.y, g.z, g.w, garea);
    if (iou > ovbest) { ovbest = iou; ovarg = m; }     // first-max semantics

    unsigned w = (m < 32) ? (mb0 >> m) : (m < 64) ? (mb1 >> (m - 32)) : (mb2 >> (m - 64));
    bool bit = (w & 1u) != 0u;
    float mg = smg[m];
    bool mp = in_gts && bit && (mg > 0.f);
    if (mp) { ++fgc; if (firstm < 0) firstm = m; }

    float s = srow[sgl[m]];
    float sig = 1.f / (1.f + expf(-s));
    float i2 = iou * iou;
    float al = sig * (i2 * i2 * i2) * mg;
    na = fmaxf(na, al / smx[m]);                       // smx >= EPS
  }

  bool fg; int tgt;
  if (fgc > 1)       { fg = true;  tgt = ovarg;  }     // multi -> IoU argmax one-hot
  else if (fgc == 1) { fg = true;  tgt = firstm; }     // argmax of 0/1 = first set
  else               { fg = false; tgt = 0;      }

  float nafin = fg ? na : 0.f;
  int lbl = sgl[tgt]; lbl = lbl < 0 ? 0 : (lbl > CC - 1 ? CC - 1 : lbl);
  float tval = fg ? (float)((int)nafin) : 0.f;         // int32 truncation per reference

  if (valid) {
    reinterpret_cast<float4*>(out_bb)[(size_t)b * NN + n] = sgt4[tgt];  // b128 store
    out_fg[(size_t)b * NN + n] = fg ? 1.f : 0.f;
    out_ti[(size_t)b * NN + n] = (float)tgt;
  }

  // ---- coalesced one-hot score write via LDS relay, b128 stores ----
  __shared__ int   slbl[TPB];
  __shared__ float sval[TPB];
  slbl[t] = lbl; sval[t] = tval;
  __syncthreads();

  int vcnt = NN - n0; if (vcnt > TPB) vcnt = TPB;
  float4* base4 = reinterpret_cast<float4*>(out_sc + ((size_t)b * NN + n0) * CC);
  const int QPA = CC / 4;  // 20 float4 per anchor
  for (int e = t; e < vcnt * QPA; e += TPB) {
    int al_ = e / QPA;
    int c0  = (e - al_ * QPA) * 4;
    int lb  = slbl[al_];
    float v = sval[al_];
    float4 o;
    o.x = (c0 + 0 == lb) ? v : 0.f;
    o.y = (c0 + 1 == lb) ? v : 0.f;
    o.z = (c0 + 2 == lb) ? v : 0.f;
    o.w = (c0 + 3 == lb) ? v : 0.f;
    base4[e] = o;
  }
}

// ---------------- launch ----------------
extern "C" void kernel_launch(void* const* d_in, const int* in_sizes, int n_in,
                              void* d_out, int out_size, void* d_ws, size_t ws_size,
                              hipStream_t stream) {
  const float* pd_scores     = (const float*)d_in[0];
  const float* pd_bboxes     = (const float*)d_in[1];
  const float* anchor_points = (const float*)d_in[2];
  const int*   gt_labels     = (const int*)d_in[3];
  const float* gt_bboxes     = (const float*)d_in[4];
  const float* mask_gt       = (const float*)d_in[5];

  char* wsb = (char*)d_ws;
  float*    maxpg    = (float*)wsb;                 // B*M floats
  unsigned* maskbits = (unsigned*)(wsb + 8192);     // B*N*3 u32

  float* out = (float*)d_out;
  float* out_bb = out;                              // B*N*4
  float* out_sc = out + (size_t)BB * NN * 4;        // B*N*C
  float* out_fg = out_sc + (size_t)BB * NN * CC;    // B*N
  float* out_ti = out_fg + (size_t)BB * NN;         // B*N

  const int nbits = BB * NN * 3;
  taa_zero_bits<<<(nbits + TPB - 1) / TPB, TPB, 0, stream>>>(maskbits, nbits);

  taa_topk_kernel<<<BB * MM, TPB, 0, stream>>>(
      pd_scores, pd_bboxes, gt_labels, gt_bboxes, mask_gt, maxpg, maskbits);

  dim3 grid((NN + TPB - 1) / TPB, BB);
  taa_assign_kernel<<<grid, TPB, 0, stream>>>(
      pd_scores, pd_bboxes, anchor_points, gt_labels, gt_bboxes, mask_gt,
      maxpg, maskbits, out_bb, out_sc, out_fg, out_ti);
}